// Correction_Module_dense_13615046328576
// MI455X (gfx1250) — compile-verified
//
#include <hip/hip_runtime.h>
#include <hip/hip_bf16.h>
#include <stdint.h>

typedef float f4 __attribute__((ext_vector_type(4)));

#define THREADS 256
#define COLS_PER_BLOCK 1024   // THREADS * 4 floats
#define ROWS_PER_BLOCK 8

__device__ __forceinline__ float cleanf(float x) {
    // zero out NaN and +/-Inf (exponent all-ones)
    unsigned u = __float_as_uint(x);
    return ((u & 0x7f800000u) == 0x7f800000u) ? 0.0f : x;
}

// Precompute lower/upper = mean -/+ k*sqrt(var) into workspace.
__global__ void bounds_kernel(const float* __restrict__ mean,
                              const float* __restrict__ var,
                              const int* __restrict__ kptr,
                              float* __restrict__ lower,
                              float* __restrict__ upper,
                              int N) {
    int i = blockIdx.x * blockDim.x + threadIdx.x;
    if (i < N) {
        float k = (float)kptr[0];
        float s = sqrtf(var[i]);
        float m = mean[i];
        lower[i] = m - k * s;
        upper[i] = m + k * s;
    }
}

__global__ __launch_bounds__(THREADS) void correction_kernel(
    const float* __restrict__ src,
    const float* __restrict__ lowerG,
    const float* __restrict__ upperG,
    float* __restrict__ dst,
    int N, int B) {
    __shared__ float sLower[COLS_PER_BLOCK];   // 4 KB
    __shared__ float sUpper[COLS_PER_BLOCK];   // 4 KB

    const int tid     = threadIdx.x;
    const int colBase = blockIdx.x * COLS_PER_BLOCK;
    const int col     = colBase + tid * 4;

    // ---- Stage this block's column bounds into LDS with gfx1250 async
    // global->LDS b128 copies (tracked by ASYNCcnt). 256 lanes x 16B = 4KB
    // per array, one instruction each per lane.
    {
        const char* gl = (const char*)(lowerG + colBase) + tid * 16;
        const char* gu = (const char*)(upperG + colBase) + tid * 16;
        // Low 32 bits of a flat shared address == LDS byte offset (aperture
        // occupies bits 63:32 on CDNA5).
        unsigned ll = (unsigned)(size_t)(void*)sLower + (unsigned)(tid * 16);
        unsigned lu = (unsigned)(size_t)(void*)sUpper + (unsigned)(tid * 16);
        asm volatile("global_load_async_to_lds_b128 %0, %1, off"
                     :: "v"(ll), "v"(gl) : "memory");
        asm volatile("global_load_async_to_lds_b128 %0, %1, off"
                     :: "v"(lu), "v"(gu) : "memory");
        asm volatile("s_wait_asynccnt 0" ::: "memory");
    }
    __syncthreads();

    if (col + 3 >= N) return;

    // Bounds for this thread's 4 fixed columns live in registers for all rows.
    f4 lo = *(const f4*)&sLower[tid * 4];
    f4 hi = *(const f4*)&sUpper[tid * 4];

    const int rowStart = blockIdx.y * ROWS_PER_BLOCK;
    const int rowStop  = (rowStart + ROWS_PER_BLOCK < B) ? (rowStart + ROWS_PER_BLOCK) : B;

    const int prevCol = (col == 0) ? (N - 1) : (col - 1);

    #pragma unroll 2
    for (int row = rowStart; row < rowStop; ++row) {
        const float* rp = src + (size_t)row * N;
        float*       wp = dst + (size_t)row * N;

        // Prefetch next row's tile while this one is in flight (global_prefetch_b8).
        if (row + 1 < rowStop) {
            __builtin_prefetch(src + (size_t)(row + 1) * N + col, 0, 3);
        }

        f4    x  = *(const f4*)(rp + col);    // global_load_b128
        float pv = rp[prevCol];               // neighbor: same/adjacent line, cache hit

        float x0 = cleanf(x.x), x1 = cleanf(x.y), x2 = cleanf(x.z), x3 = cleanf(x.w);
        float p  = cleanf(pv);

        float g0 = x0 - p;
        float g1 = x1 - x0;
        float g2 = x2 - x1;
        float g3 = x3 - x2;

        f4 r;
        r.x = (g0 < lo.x || g0 > hi.x) ? 0.0f : x0;
        r.y = (g1 < lo.y || g1 > hi.y) ? 0.0f : x1;
        r.z = (g2 < lo.z || g2 > hi.z) ? 0.0f : x2;
        r.w = (g3 < lo.w || g3 > hi.w) ? 0.0f : x3;

        // Write-once stream: non-temporal 128-bit store.
        __builtin_nontemporal_store(r, (f4*)(wp + col));
    }
}

extern "C" void kernel_launch(void* const* d_in, const int* in_sizes, int n_in,
                              void* d_out, int out_size, void* d_ws, size_t ws_size,
                              hipStream_t stream) {
    const float* output    = (const float*)d_in[0];
    const float* mean_grad = (const float*)d_in[1];
    const float* var_grad  = (const float*)d_in[2];
    const int*   kptr      = (const int*)d_in[3];

    const int N = in_sizes[1];          // 8192
    const int B = in_sizes[0] / N;      // 4096

    float* lower = (float*)d_ws;        // N floats
    float* upper = lower + N;           // N floats  (64 KB total workspace)

    bounds_kernel<<<(N + 255) / 256, 256, 0, stream>>>(
        mean_grad, var_grad, kptr, lower, upper, N);

    dim3 grid((N + COLS_PER_BLOCK - 1) / COLS_PER_BLOCK,
              (B + ROWS_PER_BLOCK - 1) / ROWS_PER_BLOCK);   // 8 x 512 = 4096 blocks
    correction_kernel<<<grid, THREADS, 0, stream>>>(
        output, lower, upper, (float*)d_out, N, B);
}